// RGCNModel_20383914787482
// MI455X (gfx1250) — compile-verified
//
#include <hip/hip_runtime.h>
#include <math.h>

#define N_NODES 50000
#define N_EDGES 800000
#define D_IN    128
#define D_OUT   128
#define N_RELS  40
#define GAMMA_F 12.0f
#define PHASE_W 0.5f
#define LN_EPS  1e-5f
#define PI_F    3.14159265358979323846f

// C row layout: [0,256) = hb (x@bases0 | x@bases1), [256,384) = xa, [384,512) = xl
#define LDC 512

typedef __attribute__((ext_vector_type(2))) float v2f;
typedef __attribute__((ext_vector_type(8))) float v8f;

// ---------------------------------------------------------------- utilities
__device__ __forceinline__ unsigned f2ord(float f) {
    unsigned u = __float_as_uint(f);
    return (u & 0x80000000u) ? ~u : (u | 0x80000000u);
}
__device__ __forceinline__ float ord2f(unsigned u) {
    unsigned b = (u & 0x80000000u) ? (u & 0x7FFFFFFFu) : ~u;
    return __uint_as_float(b);
}

// ---------------------------------------------------------------- K0: zero
__global__ void k_zero(float* __restrict__ p, int n) {
    int i = blockIdx.x * blockDim.x + threadIdx.x;
    if (i < n) p[i] = 0.0f;
}

// ---------------------------------------------------------------- K0b: pack B = [bases0 | bases1 | attn_w | loop_w] as [128][512]
__global__ void k_pack_B(const float* __restrict__ bases,
                         const float* __restrict__ attn_w,
                         const float* __restrict__ loop_w,
                         float* __restrict__ Bcat)
{
    const int i = blockIdx.x * blockDim.x + threadIdx.x;   // 128*512
    if (i >= D_IN * LDC) return;
    const int krow = i >> 9;
    const int col  = i & (LDC - 1);
    float v;
    if (col < 128)      v = bases[krow * 128 + col];
    else if (col < 256) v = bases[128 * 128 + krow * 128 + (col - 128)];
    else if (col < 384) v = attn_w[krow * 128 + (col - 256)];
    else                v = loop_w[krow * 128 + (col - 384)];
    Bcat[i] = v;
}

// ---------------------------------------------------------------- K1: fused node GEMM
// C[50000 x 512] = x[50000 x 128] @ Bcat[128 x 512], fp32 WMMA 16x16x4.
// Block = 256 threads = 8 waves, 64-row M tile (4 sub-tiles of 16).
// Wave w owns columns [64w, 64w+64) = 4 col-tiles. 16 accumulators/wave;
// each B fragment load is reused by 4 WMMAs (one per M sub-tile).
__global__ __launch_bounds__(256)
void k_gemm_node(const float* __restrict__ x,
                 const float* __restrict__ Bcat,
                 float* __restrict__ C)
{
    const int lane  = threadIdx.x & 31;
    const int wave  = threadIdx.x >> 5;
    const int half  = lane >> 4;        // 0: lanes 0-15 (K0/K1), 1: lanes 16-31 (K2/K3)
    const int l15   = lane & 15;
    const int m0    = blockIdx.x * 64;
    const int gcol0 = wave * 64;        // first column of this wave's 4 tiles

    // A row pointers per M sub-tile (clamped for the one partial block)
    const float* arow[4];
#pragma unroll
    for (int s = 0; s < 4; ++s) {
        int row = m0 + 16 * s + l15;
        if (row >= N_NODES) row = N_NODES - 1;
        arow[s] = x + (size_t)row * D_IN;
    }

    const float* bbase = Bcat + gcol0 + l15;

    v8f acc[4][4] = {};                 // [col-tile][M sub-tile]

    for (int k = 0; k < D_IN; k += 4) {
        const int kk = k + half * 2;
        // A fragments (16x4 fp32): lanes 0-15 hold K=k..k+1, lanes 16-31 K=k+2..k+3
        v2f a[4];
#pragma unroll
        for (int s = 0; s < 4; ++s)
            a[s] = *(const v2f*)(arow[s] + kk);
#pragma unroll
        for (int j = 0; j < 4; ++j) {
            // B fragment (4x16): row K striped across lanes within a VGPR
            v2f b;
            b.x = bbase[(size_t)kk * LDC + j * 16];
            b.y = bbase[(size_t)(kk + 1) * LDC + j * 16];
#pragma unroll
            for (int s = 0; s < 4; ++s)
                acc[j][s] = __builtin_amdgcn_wmma_f32_16x16x4_f32(
                    false, a[s], false, b, (short)0, acc[j][s], false, false);
        }
    }

    // C/D layout: VGPR r -> M = r + 8*half, N = lane&15
#pragma unroll
    for (int s = 0; s < 4; ++s) {
        if (m0 + 16 * s < N_NODES) {
            float* crow = C + (size_t)(m0 + 16 * s + half * 8) * LDC + gcol0 + l15;
#pragma unroll
            for (int j = 0; j < 4; ++j)
#pragma unroll
                for (int r = 0; r < 8; ++r)
                    crow[(size_t)r * LDC + j * 16] = acc[j][s][r];
        }
    }
}

// ---------------------------------------------------------------- K2: s2[r] = m_rel @ (attn_w @ a2)
__global__ void k_rel_path(const float* __restrict__ m_rel,
                           const float* __restrict__ attn_w,
                           const float* __restrict__ attn_vec,
                           float* __restrict__ s2)
{
    __shared__ float waw2[D_IN];
    const int t = threadIdx.x;           // blockDim = 128
    const float* a2 = attn_vec + D_IN;
    float acc = 0.0f;
    for (int o = 0; o < D_IN; ++o) acc += attn_w[t * D_IN + o] * a2[o];
    waw2[t] = acc;
    __syncthreads();
    if (t < N_RELS) {
        float s = 0.0f;
        for (int i = 0; i < D_IN; ++i) s += m_rel[t * D_IN + i] * waw2[i];
        s2[t] = s;
    }
}

// ---------------------------------------------------------------- K3: s1/s3 per node (one wave per node); xa = C row + 256
__global__ __launch_bounds__(256)
void k_node_scores(const float* __restrict__ C,
                   const float* __restrict__ attn_vec,
                   float* __restrict__ s1, float* __restrict__ s3)
{
    const int n    = (blockIdx.x * blockDim.x + threadIdx.x) >> 5;
    const int lane = threadIdx.x & 31;
    if (n >= N_NODES) return;
    const float4 v  = ((const float4*)(C + (size_t)n * LDC + 256))[lane];
    const float4 a1 = ((const float4*)(attn_vec))[lane];
    const float4 a3 = ((const float4*)(attn_vec + 2 * D_IN))[lane];
    float d1 = v.x * a1.x + v.y * a1.y + v.z * a1.z + v.w * a1.w;
    float d3 = v.x * a3.x + v.y * a3.y + v.z * a3.z + v.w * a3.w;
#pragma unroll
    for (int m = 16; m >= 1; m >>= 1) {
        d1 += __shfl_xor(d1, m, 32);
        d3 += __shfl_xor(d3, m, 32);
    }
    if (lane == 0) { s1[n] = d1; s3[n] = d3; }
}

// ---------------------------------------------------------------- K4: logits + segment max (monotonic-uint atomicMax)
__global__ void k_logit(const int* __restrict__ src, const int* __restrict__ dst,
                        const int* __restrict__ et,
                        const float* __restrict__ s1, const float* __restrict__ s2,
                        const float* __restrict__ s3,
                        float* __restrict__ logit, unsigned* __restrict__ maxk)
{
    const int e = blockIdx.x * blockDim.x + threadIdx.x;
    if (e >= N_EDGES) return;
    const float l = s1[src[e]] + s2[et[e]] + s3[dst[e]];
    logit[e] = l;
    atomicMax(maxk + dst[e], f2ord(l));
}

// ---------------------------------------------------------------- K5: exp + segment sum
__global__ void k_exp(const int* __restrict__ dst, const float* __restrict__ logit,
                      const unsigned* __restrict__ maxk,
                      float* __restrict__ ex, float* __restrict__ den)
{
    const int e = blockIdx.x * blockDim.x + threadIdx.x;
    if (e >= N_EDGES) return;
    const int d = dst[e];
    const float v = expf(logit[e] - ord2f(maxk[d]));
    ex[e] = v;
    atomicAdd(den + d, v);
}

// ---------------------------------------------------------------- K6: weighted message scatter (one wave per edge)
// hb rows live in C at stride 512 (cols 0..255); gathers hit L2 (C is L2-resident hot set).
__global__ __launch_bounds__(256)
void k_scatter(const int* __restrict__ src, const int* __restrict__ dst,
               const int* __restrict__ et, const float* __restrict__ norm,
               const float* __restrict__ coef, const float* __restrict__ C,
               const float* __restrict__ ex, const float* __restrict__ den,
               float* __restrict__ h)
{
    const int e    = (blockIdx.x * blockDim.x + threadIdx.x) >> 5;
    const int lane = threadIdx.x & 31;
    if (e >= N_EDGES) return;
    const int s = src[e], d = dst[e], r = et[e];
    const float w  = ex[e] / den[d] * norm[e];
    const float c0 = coef[r * 2 + 0] * w;
    const float c1 = coef[r * 2 + 1] * w;
    const float4 v0 = ((const float4*)(C + (size_t)s * LDC))[lane];
    const float4 v1 = ((const float4*)(C + (size_t)s * LDC + 128))[lane];
    float* hd = h + (size_t)d * 128 + lane * 4;
    atomicAdd(hd + 0, c0 * v0.x + c1 * v1.x);
    atomicAdd(hd + 1, c0 * v0.y + c1 * v1.y);
    atomicAdd(hd + 2, c0 * v0.z + c1 * v1.z);
    atomicAdd(hd + 3, c0 * v0.w + c1 * v1.w);
}

// ---------------------------------------------------------------- K7: LayerNorm + bias + loop-proj + polar split (one wave per node)
__global__ __launch_bounds__(256)
void k_finalize(const float* __restrict__ h, const float* __restrict__ C,
                const float* __restrict__ h_bias, const float* __restrict__ ln_g,
                const float* __restrict__ ln_b, const float* __restrict__ negat,
                float* __restrict__ out)
{
    const int n    = (blockIdx.x * blockDim.x + threadIdx.x) >> 5;
    const int lane = threadIdx.x & 31;
    if (n >= N_NODES) return;
    const float4 v = ((const float4*)(h + (size_t)n * 128))[lane];
    float sum = v.x + v.y + v.z + v.w;
#pragma unroll
    for (int m = 16; m >= 1; m >>= 1) sum += __shfl_xor(sum, m, 32);
    const float mu = sum * (1.0f / 128.0f);
    float4 c = make_float4(v.x - mu, v.y - mu, v.z - mu, v.w - mu);
    float ss = c.x * c.x + c.y * c.y + c.z * c.z + c.w * c.w;
#pragma unroll
    for (int m = 16; m >= 1; m >>= 1) ss += __shfl_xor(ss, m, 32);
    const float rstd = rsqrtf(ss * (1.0f / 128.0f) + LN_EPS);

    const float4 g   = ((const float4*)ln_g)[lane];
    const float4 b   = ((const float4*)ln_b)[lane];
    const float4 hb4 = ((const float4*)h_bias)[lane];
    const float4 x4  = ((const float4*)(C + (size_t)n * LDC + 384))[lane]; // xl
    float4 hv;
    hv.x = c.x * rstd * g.x + b.x + hb4.x + x4.x;
    hv.y = c.y * rstd * g.y + b.y + hb4.y + x4.y;
    hv.z = c.z * rstd * g.z + b.z + hb4.z + x4.z;
    hv.w = c.w * rstd * g.w + b.w + hb4.w + x4.w;

    // Channel c on lane l pairs with channel c+64 on lane l^16.
    float4 ov;
    ov.x = __shfl_xor(hv.x, 16, 32);
    ov.y = __shfl_xor(hv.y, 16, 32);
    ov.z = __shfl_xor(hv.z, 16, 32);
    ov.w = __shfl_xor(hv.w, 16, 32);

    float4 res;
    if (lane < 16) {  // modulus half: hv = x_x, ov = x_y
        const float4 ng = ((const float4*)(negat + (size_t)n * 64))[lane];
        res.x = sqrtf(hv.x * hv.x + ov.x * ov.x) * ng.x;
        res.y = sqrtf(hv.y * hv.y + ov.y * ov.y) * ng.y;
        res.z = sqrtf(hv.z * hv.z + ov.z * ov.z) * ng.z;
        res.w = sqrtf(hv.w * hv.w + ov.w * ov.w) * ng.w;
    } else {          // phase half: hv = x_y, ov = x_x
        res.x = atan2f(hv.x, ov.x) + PI_F;
        res.y = atan2f(hv.y, ov.y) + PI_F;
        res.z = atan2f(hv.z, ov.z) + PI_F;
        res.w = atan2f(hv.w, ov.w) + PI_F;
    }
    ((float4*)(out + (size_t)n * 128))[lane] = res;
}

// ---------------------------------------------------------------- K8: HAKE score (one wave per triple)
__global__ __launch_bounds__(256)
void k_hake(const int* __restrict__ triples, const float* __restrict__ out,
            const float* __restrict__ hake_rel, float* __restrict__ score)
{
    const int t    = (blockIdx.x * blockDim.x + threadIdx.x) >> 5;
    const int lane = threadIdx.x & 31;
    if (t >= 1024 * 32) return;
    const int hI = triples[t * 3 + 0];
    const int r  = triples[t * 3 + 1];
    const int tI = triples[t * 3 + 2];
    const float2* H = (const float2*)(out + (size_t)hI * 128);
    const float2* T = (const float2*)(out + (size_t)tI * 128);
    const float2* R = (const float2*)(hake_rel + (size_t)r * 128);
    const float2 hm = H[lane],      tm = T[lane],      rm = R[lane];
    const float2 hp = H[lane + 32], tp = T[lane + 32], rp = R[lane + 32];
    const float d0 = hm.x * rm.x - tm.x;
    const float d1 = hm.y * rm.y - tm.y;
    float mod = d0 * d0 + d1 * d1;
    float ph  = fabsf(sinf((hp.x + rp.x - tp.x) * 0.5f)) +
                fabsf(sinf((hp.y + rp.y - tp.y) * 0.5f));
#pragma unroll
    for (int m = 16; m >= 1; m >>= 1) {
        mod += __shfl_xor(mod, m, 32);
        ph  += __shfl_xor(ph, m, 32);
    }
    if (lane == 0)
        score[t] = GAMMA_F - (sqrtf(mod + 1e-12f) + PHASE_W * ph);
}

// ---------------------------------------------------------------- launch
extern "C" void kernel_launch(void* const* d_in, const int* in_sizes, int n_in,
                              void* d_out, int out_size, void* d_ws, size_t ws_size,
                              hipStream_t stream)
{
    const int*   src       = (const int*)d_in[0];
    const int*   dst       = (const int*)d_in[1];
    const int*   etypes    = (const int*)d_in[2];
    const float* norm      = (const float*)d_in[3];
    const int*   triples   = (const int*)d_in[4];
    const float* emb       = (const float*)d_in[5];
    const float* bases     = (const float*)d_in[6];
    const float* coef      = (const float*)d_in[7];
    const float* h_bias    = (const float*)d_in[8];
    const float* ln_gamma  = (const float*)d_in[9];
    const float* ln_beta   = (const float*)d_in[10];
    const float* loop_w    = (const float*)d_in[11];
    const float* attn_w    = (const float*)d_in[12];
    const float* attn_vec  = (const float*)d_in[13];
    const float* m_rel     = (const float*)d_in[14];
    const float* negat     = (const float*)d_in[15];
    const float* hake_rel  = (const float*)d_in[16];

    float* W = (float*)d_ws;
    size_t off = 0;
    float* Bcat  = W + off; off += (size_t)D_IN * LDC;      // packed weights [128][512]
    float* C     = W + off; off += (size_t)N_NODES * LDC;   // hb | xa | xl per row
    float* s1    = W + off; off += N_NODES;
    float* s3    = W + off; off += N_NODES;
    float* s2    = W + off; off += 64;
    float* logit = W + off; off += N_EDGES;
    float* exb   = W + off; off += N_EDGES;
    float* h     = W + off; off += (size_t)N_NODES * 128;   // scatter target
    float* den   = W + off; off += N_NODES;                 // scatter target
    unsigned* maxk = (unsigned*)(W + off); off += N_NODES;  // scatter target
    float* outn  = W + off; off += (size_t)N_NODES * 128;

    // zero the scatter targets (h, den, maxk are contiguous: N*130 words)
    {
        const int nz = N_NODES * 130;
        k_zero<<<(nz + 255) / 256, 256, 0, stream>>>(h, nz);
    }

    k_pack_B<<<(D_IN * LDC + 255) / 256, 256, 0, stream>>>(bases, attn_w, loop_w,
                                                           Bcat);
    k_gemm_node<<<(N_NODES + 63) / 64, 256, 0, stream>>>(emb, Bcat, C);
    k_rel_path<<<1, 128, 0, stream>>>(m_rel, attn_w, attn_vec, s2);
    k_node_scores<<<(N_NODES * 32 + 255) / 256, 256, 0, stream>>>(C, attn_vec,
                                                                  s1, s3);
    k_logit<<<(N_EDGES + 255) / 256, 256, 0, stream>>>(src, dst, etypes,
                                                       s1, s2, s3, logit, maxk);
    k_exp<<<(N_EDGES + 255) / 256, 256, 0, stream>>>(dst, logit, maxk, exb, den);
    k_scatter<<<((size_t)N_EDGES * 32 + 255) / 256, 256, 0, stream>>>(
        src, dst, etypes, norm, coef, C, exb, den, h);
    k_finalize<<<(N_NODES * 32 + 255) / 256, 256, 0, stream>>>(
        h, C, h_bias, ln_gamma, ln_beta, negat, outn);
    k_hake<<<(1024 * 32 * 32 + 255) / 256, 256, 0, stream>>>(
        triples, outn, hake_rel, (float*)d_out);
}